// photoreceptor_DA_multichan_norm_75101798138141
// MI455X (gfx1250) — compile-verified
//
#include <hip/hip_runtime.h>
#include <math.h>

typedef __attribute__((ext_vector_type(2))) float v2f;
typedef __attribute__((ext_vector_type(4))) float v4f;
typedef __attribute__((ext_vector_type(8))) float v8f;

// Problem constants (from reference): B=65536, T=64, U=8
#define BN 65536
#define TN 64
#define UN 8

// d_ws float layout:
//   [0    .. 1023] : Bmat[k][n] = W[n][k]  (k=0..63, n=0..15; n>=8 are zero pad)
//   [1024 .. 1031] : acoef[u] = alpha[u]*100
//   [1032 .. 1039] : bcoef[u] = beta[u]*10

// ---------------------------------------------------------------------------
// Prep kernel: build Kz (64x8), per-channel argmax -> t_idx, then the padded
// 64x16 weight matrix for the WMMA B operand.  One block of 64 threads.
// ---------------------------------------------------------------------------
__global__ void pr_prep(const float* __restrict__ alpha,
                        const float* __restrict__ beta,
                        const float* __restrict__ gamma,
                        const float* __restrict__ tauZ,
                        const float* __restrict__ nZ,
                        const float* __restrict__ tauC,
                        const float* __restrict__ nC,
                        float* __restrict__ ws)
{
    __shared__ float kz[TN * UN];
    __shared__ int   tidx[UN];

    const int t = threadIdx.x;          // 0..63
    const float ft = (float)t;

    for (int u = 0; u < UN; ++u) {
        float tc  = tauC[u] * 100.0f;
        float ncv = nC[u]   * 10.0f;
        float tz  = tauZ[u] * 100.0f;
        float nzv = nZ[u]   * 10.0f;
        float g   = gamma[u] * 10.0f;
        float kc = powf(ft, ncv) * expf(-ft / tc) /
                   (powf(tc, ncv + 1.0f) * expf(lgammaf(ncv + 1.0f)));
        float kf = powf(ft, nzv) * expf(-ft / tz) /
                   (powf(tz, nzv + 1.0f) * expf(lgammaf(nzv + 1.0f)));
        kz[t * UN + u] = g * kc + (1.0f - g) * kf;
    }
    __syncthreads();

    if (t < UN) {
        // argmax over time axis (first occurrence of max, like jnp.argmax)
        float best = kz[0 * UN + t];
        int bi = 0;
        for (int i = 1; i < TN; ++i) {
            float v = kz[i * UN + t];
            if (v > best) { best = v; bi = i; }
        }
        int ti = TN - bi;
        ti = ti < 0 ? 0 : (ti > TN - 1 ? TN - 1 : ti);
        tidx[t] = ti;
        ws[1024 + t] = alpha[t] * 100.0f;
        ws[1032 + t] = beta[t]  * 10.0f;
    }
    __syncthreads();

    // Bmat[k=t][n] = (t <= tidx[n]) ? Kz[tidx[n]-t, n] : 0   (n>=8 -> 0)
    for (int n = 0; n < 16; ++n) {
        float v = 0.0f;
        if (n < UN) {
            int ti = tidx[n];
            if (t <= ti) v = kz[(ti - t) * UN + n];
        }
        ws[t * 16 + n] = v;
    }
}

// ---------------------------------------------------------------------------
// Main kernel: per wave, one 16-row tile.
//   Phase 1: z_sel tile = X(16x64) @ Bmat(64x16) via 16 chained
//            v_wmma_f32_16x16x4_f32 (full fp32).
//   Phase 2: cc[row][u] = a[u]/(1+b[u]*z) staged through LDS.
//   Phase 3: stream out[row,t,u] = x[row,t]*cc[row][u] with NT float4 stores.
// ---------------------------------------------------------------------------
__global__ __launch_bounds__(256) void pr_main(const float* __restrict__ x,
                                               const float* __restrict__ ws,
                                               float* __restrict__ out)
{
    const int lane = threadIdx.x & 31;
    const int wave = threadIdx.x >> 5;
    const int tile = blockIdx.x * 8 + wave;      // 16 rows per tile
    const int n    = lane & 15;                   // M for A, N for B/C/D
    const int h    = lane >> 4;                   // half-wave select

    const float* xrowA = x + (size_t)(tile * 16 + n) * TN;

    // Load all 16 B fragments (shared across k-steps; 64x16 fp32 in ws).
    // B layout: lane h*16+n, VGPR j holds B[4*kb + 2*h + j][n].
    v2f bf[16];
#pragma unroll
    for (int kb = 0; kb < 16; ++kb) {
        int k0 = 4 * kb + 2 * h;
        v2f b;
        b[0] = ws[(k0 + 0) * 16 + n];
        b[1] = ws[(k0 + 1) * 16 + n];
        bf[kb] = b;
    }

    // A layout: lane h*16+m, VGPR j holds A[m][4*kb + 2*h + j]  (fp32 x data)
    v8f c = {};
#pragma unroll
    for (int kb = 0; kb < 16; ++kb) {
        v2f a = *(const v2f*)(xrowA + 4 * kb + 2 * h);
        c = __builtin_amdgcn_wmma_f32_16x16x4_f32(
                false, a, false, bf[kb], (short)0, c, false, false);
    }

    // D layout: lane h*16+n, VGPR r holds D[row = r + 8h][n]
    __shared__ float lds[8 * 16 * UN];            // 16 rows x 8 chans per wave
    float* myl = lds + wave * (16 * UN);
    if (n < UN) {
        float ac = ws[1024 + n];
        float bc = ws[1032 + n];
#pragma unroll
        for (int r = 0; r < 8; ++r) {
            float z = c[r];
            myl[(r + 8 * h) * UN + n] = ac / (1.0f + bc * z);
        }
    }
    __syncthreads();

    // Streaming output: per row, 512 floats = 128 float4; 32 lanes x 4 iters.
    const int par = lane & 1;                     // which half of the 8 chans
    for (int r16 = 0; r16 < 16; ++r16) {
        int row = tile * 16 + r16;
        v4f cv = *(const v4f*)(myl + r16 * UN + par * 4);
        const float* xr  = x + (size_t)row * TN;
        float* orow      = out + (size_t)row * (TN * UN);
#pragma unroll
        for (int i = 0; i < 4; ++i) {
            int f = i * 32 + lane;                // float4 index within row
            int t = f >> 1;
            float xv = xr[t];
            v4f o = xv * cv;
            __builtin_nontemporal_store(o, (v4f*)(orow + 4 * f));
        }
    }
}

// ---------------------------------------------------------------------------
extern "C" void kernel_launch(void* const* d_in, const int* in_sizes, int n_in,
                              void* d_out, int out_size, void* d_ws, size_t ws_size,
                              hipStream_t stream)
{
    (void)in_sizes; (void)n_in; (void)out_size; (void)ws_size;
    // setup_inputs order: inputs, alpha, beta, gamma, zeta, tauY, nY,
    //                     tauZ, nZ, tauC, nC
    const float* x     = (const float*)d_in[0];
    const float* alpha = (const float*)d_in[1];
    const float* beta  = (const float*)d_in[2];
    const float* gamma = (const float*)d_in[3];
    const float* tauZ  = (const float*)d_in[7];
    const float* nZ    = (const float*)d_in[8];
    const float* tauC  = (const float*)d_in[9];
    const float* nC    = (const float*)d_in[10];
    float*       out   = (float*)d_out;
    float*       ws    = (float*)d_ws;

    pr_prep<<<1, 64, 0, stream>>>(alpha, beta, gamma, tauZ, nZ, tauC, nC, ws);

    // B/16 = 4096 tiles, 8 waves (tiles) per 256-thread block -> 512 blocks
    pr_main<<<BN / (16 * 8), 256, 0, stream>>>(x, ws, out);
}